// CombinedPooling_49435073577305
// MI455X (gfx1250) — compile-verified
//
#include <hip/hip_runtime.h>

#define NUM_OUTN 8192
#define C_IN 128
#define CONCATN 8320
#define B_ 8
#define T_ 8
#define HW_ 196
#define NPOS 10976   // 7*8*196
#define MID1N 4096
#define MID2N 1024
#define EPSV 1e-5f

typedef float v2f __attribute__((ext_vector_type(2)));
typedef float v8f __attribute__((ext_vector_type(8)));

__global__ void kzero(float* p, int n) {
  int i = blockIdx.x * blockDim.x + threadIdx.x;
  if (i < n) p[i] = 0.f;
}

// Extract count-sketch (index, sign) from dense P1/P2 rows (one nonzero each).
__global__ void ksketch(const float* __restrict__ P1, const float* __restrict__ P2,
                        int* h1, float* s1, int* h2, float* s2) {
  int tid = threadIdx.x;
  const float* row; int* hp; float* sp; int c;
  if (tid < C_IN) { c = tid;        row = P1 + (size_t)c * NUM_OUTN; hp = h1; sp = s1; }
  else            { c = tid - C_IN; row = P2 + (size_t)c * NUM_OUTN; hp = h2; sp = s2; }
  int idx = 0; float val = 0.f;
  for (int k = 0; k < NUM_OUTN; ++k) {
    float v = row[k];
    if (v != 0.f) { idx = k; val = v; }
  }
  hp[c] = idx; sp[c] = val;
}

// One block per (t,b,i,j) position: 128x128 outer-product scatter into LDS,
// then dump 8192 bilinear channels + 128 dual-avg channels to raw[pos][8320].
__global__ void kbilinear(const float* __restrict__ x,
                          const int* __restrict__ h1, const float* __restrict__ s1,
                          const int* __restrict__ h2, const float* __restrict__ s2,
                          float* __restrict__ raw) {
  __shared__ float out[NUM_OUTN];
  __shared__ float q1[C_IN], q2[C_IN];
  __shared__ int hh1[C_IN], hh2[C_IN];
  int tid = threadIdx.x;
  int p = blockIdx.x;                 // ((t*8 + b)*196 + ij)
  int t = p / (B_ * HW_);
  int rem = p - t * (B_ * HW_);
  int b = rem / HW_;
  int ij = rem - b * HW_;

  if (tid < C_IN) {
    int c = tid;
    float xv = x[(((size_t)b * C_IN + c) * T_ + t) * HW_ + ij];
    q1[c] = s1[c] * xv; hh1[c] = h1[c];
  } else {
    int c = tid - C_IN;
    float xv = x[(((size_t)b * C_IN + c) * T_ + (t + 1)) * HW_ + ij];
    q2[c] = s2[c] * xv; hh2[c] = h2[c];
  }
  for (int k = tid; k < NUM_OUTN; k += 256) out[k] = 0.f;
  __syncthreads();

  int c2 = tid & 127;
  float q2v = q2[c2];
  int h2v = hh2[c2];
  int c1 = tid >> 7;                  // c2 fixed per thread; c1 strides by 2
  #pragma unroll 4
  for (int it = 0; it < 64; ++it) {
    int n = (hh1[c1] + h2v) & (NUM_OUTN - 1);
    atomicAdd(&out[n], q1[c1] * q2v);
    c1 += 2;
  }
  __syncthreads();

  float* dst = raw + (size_t)p * CONCATN;
  for (int k = tid; k < NUM_OUTN; k += 256) dst[k] = out[k];
  if (tid < C_IN) {
    float x1 = q1[tid] * s1[tid];     // s = +-1, so q*s recovers x
    float x2 = q2[tid] * s2[tid];
    dst[NUM_OUTN + tid] = 0.5f * (x1 + x2);
  }
}

// Per-channel sum / sumsq over all 10976 positions (16 position slices).
__global__ void kstats1(const float* __restrict__ raw, float* chsum, float* chsq) {
  int ch = blockIdx.x * 256 + threadIdx.x;
  if (ch >= CONCATN) return;
  int p0 = blockIdx.y * (NPOS / 16);  // 686 positions per slice
  float s = 0.f, sq = 0.f;
  for (int pi = 0; pi < NPOS / 16; ++pi) {
    float v = raw[(size_t)(p0 + pi) * CONCATN + ch];
    s += v; sq += v * v;
  }
  atomicAdd(&chsum[ch], s);
  atomicAdd(&chsq[ch], sq);
}

__global__ void kstats2(const float* chsum, const float* chsq,
                        const float* gamma, const float* beta,
                        float* na, float* nb) {
  int ch = blockIdx.x * 256 + threadIdx.x;
  if (ch >= CONCATN) return;
  float mean = chsum[ch] * (1.f / NPOS);
  float var = chsq[ch] * (1.f / NPOS) - mean * mean;
  float a = gamma[ch] * rsqrtf(var + EPSV);
  na[ch] = a;
  nb[ch] = beta[ch] - mean * a;
}

// Fused: 7x7 avg pool (separable, prefix sums) -> affine norm -> ReLU ->
// global avg -> atomic accumulate into featP[b][ch].
__global__ void kpoolfeat(const float* __restrict__ raw,
                          const float* __restrict__ na, const float* __restrict__ nb,
                          float* __restrict__ featP) {
  int ch = blockIdx.x * 256 + threadIdx.x;
  if (ch >= CONCATN) return;
  int io = blockIdx.y;                 // 0..7 output row
  int tb = blockIdx.z;                 // 0..55 = t*8 + b
  int b = tb & 7;
  const float* base = raw + (size_t)tb * HW_ * CONCATN + ch;
  float colacc[8];
  #pragma unroll
  for (int jo = 0; jo < 8; ++jo) colacc[jo] = 0.f;
  for (int di = 0; di < 7; ++di) {
    int i = io + di;
    float v[14], ps[15];
    #pragma unroll
    for (int j = 0; j < 14; ++j)
      v[j] = base[(size_t)(i * 14 + j) * CONCATN];
    ps[0] = 0.f;
    #pragma unroll
    for (int j = 0; j < 14; ++j) ps[j + 1] = ps[j] + v[j];
    #pragma unroll
    for (int jo = 0; jo < 8; ++jo) colacc[jo] += ps[jo + 7] - ps[jo];
  }
  float a = na[ch], bb = nb[ch];
  float acc = 0.f;
  #pragma unroll
  for (int jo = 0; jo < 8; ++jo) {
    float val = colacc[jo] * (1.f / 49.f) * a + bb;
    acc += fmaxf(val, 0.f);
  }
  atomicAdd(&featP[(size_t)b * CONCATN + ch], acc * (1.f / 448.f));
}

// out[16,N] = relu?(A[16,K] @ W[N,K]^T + bias) via V_WMMA_F32_16X16X4_F32.
// One wave per 16-wide N tile. ISA layouts (CDNA5 ISA):
//  A 16x4: lane&15 = M, lane>>4 selects K pair {0,1}/{2,3} in the VGPR pair.
//  B 4x16: mirrored (lane&15 = N).  C/D: vgpr r -> M = r + 8*(lane>>4), N = lane&15.
__global__ void kgemm_wmma(const float* __restrict__ A, const float* __restrict__ Wt,
                           const float* __restrict__ bias, float* __restrict__ out,
                           int K, int N, int relu, int mrows) {
  int lane = threadIdx.x;
  int nl = lane & 15;
  int kh = lane >> 4;
  int n = blockIdx.x * 16 + nl;
  float bv = bias[n];
  v8f c;
  #pragma unroll
  for (int r = 0; r < 8; ++r) c[r] = bv;
  const float* arow = A + (size_t)nl * K + 2 * kh;
  const float* brow = Wt + (size_t)n * K + 2 * kh;
  for (int k = 0; k < K; k += 4) {
    v2f av = *(const v2f*)(arow + k);
    v2f bw = *(const v2f*)(brow + k);
    c = __builtin_amdgcn_wmma_f32_16x16x4_f32(false, av, false, bw,
                                              (short)0, c, false, false);
  }
  #pragma unroll
  for (int r = 0; r < 8; ++r) {
    int m = r + 8 * kh;
    if (m < mrows) {
      float v = c[r];
      if (relu) v = fmaxf(v, 0.f);
      out[(size_t)m * N + n] = v;
    }
  }
}

extern "C" void kernel_launch(void* const* d_in, const int* in_sizes, int n_in,
                              void* d_out, int out_size, void* d_ws, size_t ws_size,
                              hipStream_t stream) {
  const float* x     = (const float*)d_in[0];
  const float* P1    = (const float*)d_in[1];
  const float* P2    = (const float*)d_in[2];
  const float* gamma = (const float*)d_in[3];
  const float* beta  = (const float*)d_in[4];
  const float* W1    = (const float*)d_in[5];
  const float* b1    = (const float*)d_in[6];
  const float* W2    = (const float*)d_in[7];
  const float* b2    = (const float*)d_in[8];
  const float* W3    = (const float*)d_in[9];
  const float* b3    = (const float*)d_in[10];

  float* ws = (float*)d_ws;
  size_t off = 0;
  float* raw   = ws + off; off += (size_t)NPOS * CONCATN; // 365 MB
  float* featP = ws + off; off += 16 * CONCATN;           // [16][8320], rows 8..15 zero
  float* chsum = ws + off; off += CONCATN;
  float* chsq  = ws + off; off += CONCATN;
  float* na    = ws + off; off += CONCATN;
  float* nb    = ws + off; off += CONCATN;
  float* act1  = ws + off; off += 16 * MID1N;
  float* act2  = ws + off; off += 16 * MID2N;
  float* s1    = ws + off; off += C_IN;
  float* s2    = ws + off; off += C_IN;
  int* h1 = (int*)(ws + off); off += C_IN;
  int* h2 = (int*)(ws + off); off += C_IN;

  int nz = 16 * CONCATN + 2 * CONCATN;  // featP + chsum + chsq are contiguous
  kzero<<<(nz + 255) / 256, 256, 0, stream>>>(featP, nz);
  ksketch<<<1, 256, 0, stream>>>(P1, P2, h1, s1, h2, s2);
  kbilinear<<<NPOS, 256, 0, stream>>>(x, h1, s1, h2, s2, raw);
  kstats1<<<dim3(33, 16, 1), 256, 0, stream>>>(raw, chsum, chsq);
  kstats2<<<33, 256, 0, stream>>>(chsum, chsq, gamma, beta, na, nb);
  kpoolfeat<<<dim3(33, 8, 56), 256, 0, stream>>>(raw, na, nb, featP);
  kgemm_wmma<<<MID1N / 16, 32, 0, stream>>>(featP, W1, b1, act1, CONCATN, MID1N, 1, 16);
  kgemm_wmma<<<MID2N / 16, 32, 0, stream>>>(act1, W2, b2, act2, MID1N, MID2N, 1, 16);
  kgemm_wmma<<<128 / 16, 32, 0, stream>>>(act2, W3, b3, (float*)d_out, MID2N, 128, 0, 8);
}